// FixedRangeScaling_283467842321
// MI455X (gfx1250) — compile-verified
//
#include <hip/hip_runtime.h>

// out[b,f] = x[b,f] * weight[f]   (BATCH=16384, FEATURES=4096, fp32)
// Pure streaming op: 512 MB of HBM traffic, ~22 us floor at 23.3 TB/s.
// Strategy: b128 NT loads/stores for the 256MB x / 256MB out streams,
// temporal (cached) loads for the 16KB weight vector, 4x unroll per thread
// for latency hiding, speculative prefetch one tile ahead.

typedef float float4v __attribute__((ext_vector_type(4)));

#define FEATURES   4096
#define W4         (FEATURES / 4)          // 1024 float4 per row
#define THREADS    256
#define UNROLL     4
#define TOTAL_F4   (16384ULL * FEATURES / 4) // 16,777,216 float4 elements
#define GRID       (TOTAL_F4 / (THREADS * UNROLL)) // 16384 blocks

__global__ __launch_bounds__(THREADS)
void rowscale_f32_kernel(const float4v* __restrict__ x,
                         const float4v* __restrict__ w4,
                         float4v* __restrict__ out) {
    // Each block owns a contiguous tile of THREADS*UNROLL float4s.
    const size_t base = (size_t)blockIdx.x * (THREADS * UNROLL) + threadIdx.x;

    // Speculative prefetch of the next block's tile (safe: dropped if OOB).
    __builtin_prefetch(&x[base + (size_t)THREADS * UNROLL], 0, 0);

#pragma unroll
    for (int k = 0; k < UNROLL; ++k) {
        const size_t i4  = base + (size_t)k * THREADS;
        const int    col = (int)(i4 & (W4 - 1));     // feature column (float4 units)

        // weight: tiny (16KB), keep temporal so it lives in WGP$/L2.
        const float4v wv = w4[col];
        // x: streamed once -> non-temporal, don't pollute L2.
        const float4v xv = __builtin_nontemporal_load(&x[i4]);

        const float4v ov = xv * wv;

        // out: write-once stream -> non-temporal.
        __builtin_nontemporal_store(ov, &out[i4]);
    }
}

extern "C" void kernel_launch(void* const* d_in, const int* in_sizes, int n_in,
                              void* d_out, int out_size, void* d_ws, size_t ws_size,
                              hipStream_t stream) {
    const float4v* x  = (const float4v*)d_in[0];   // (16384, 4096) fp32
    const float4v* w4 = (const float4v*)d_in[1];   // (4096,) fp32 viewed as float4
    float4v*       o  = (float4v*)d_out;

    (void)in_sizes; (void)n_in; (void)out_size; (void)d_ws; (void)ws_size;

    rowscale_f32_kernel<<<dim3((unsigned)GRID), dim3(THREADS), 0, stream>>>(x, w4, o);
}